// TorchGradientBoostingClassifier_72000831750468
// MI455X (gfx1250) — compile-verified
//
#include <hip/hip_runtime.h>
#include <cstdint>

// ---------------- problem constants (match reference) ----------------
#define N_SAMPLES  65536
#define N_FEAT     64
#define N_TREES    600        // N_STAGES * N_CLASSES
#define N_CLASSES  3
#define TREE_DEPTH 6
#define M_NODES    63         // internal nodes per tree
#define N_LEAF     64
#define LRATE      0.1f

// ---------------- kernel tiling ----------------
#define BLOCK      128                      // 4 waves (wave32)
#define TB         12                       // trees per staged batch (multiple of 3)
#define NB         (N_TREES / TB)           // 50 batches (even -> clean 2x unroll)

// ---- CDNA5 async global->LDS helpers (gfx1250, ASYNCcnt-tracked) ----
// Low 32 bits of a flat pointer to __shared__ are the LDS byte offset
__device__ __forceinline__ unsigned lds_off(const void* p) {
    return (unsigned)(uintptr_t)p;
}
__device__ __forceinline__ void async_ld_b32(unsigned lds, const void* g) {
    asm volatile("global_load_async_to_lds_b32 %0, %1, off"
                 :: "v"(lds), "v"(g) : "memory");
}
__device__ __forceinline__ void async_ld_b128(unsigned lds, const void* g) {
    asm volatile("global_load_async_to_lds_b128 %0, %1, off"
                 :: "v"(lds), "v"(g) : "memory");
}
__device__ __forceinline__ void wait_async0() {
    asm volatile("s_wait_asynccnt 0" ::: "memory");
}

// Stage TB trees (batch starting at tree t0) into an LDS buffer.
// Node records are interleaved {feat,i32 | thresh,f32} so the traversal
// needs a single ds_load_b64 per visited node.
__device__ __forceinline__ void stage_trees(int t0, int tid,
                                            const int*   __restrict__ feat,
                                            const float* __restrict__ thresh,
                                            const float* __restrict__ leaf,
                                            uint2* __restrict__ nodesBuf,
                                            float* __restrict__ leavesBuf) {
    const int*   gf = feat   + t0 * M_NODES;
    const float* gt = thresh + t0 * M_NODES;
    unsigned nb = lds_off(nodesBuf);
    for (int e = tid; e < TB * M_NODES; e += BLOCK) {
        async_ld_b32(nb + (unsigned)e * 8u,      gf + e);   // feat  -> .x
        async_ld_b32(nb + (unsigned)e * 8u + 4u, gt + e);   // thresh-> .y
    }
    const float* gl = leaf + t0 * N_LEAF;
    unsigned lb = lds_off(leavesBuf);
    for (int e = tid; e < (TB * N_LEAF) / 4; e += BLOCK) {
        async_ld_b128(lb + (unsigned)e * 16u, gl + e * 4);
    }
}

__global__ __launch_bounds__(BLOCK)
void gbt_eval_kernel(const float* __restrict__ x,
                     const int*   __restrict__ feat,
                     const float* __restrict__ thresh,
                     const float* __restrict__ leaf,
                     float*       __restrict__ out) {
    __shared__ float xs[BLOCK * N_FEAT];           // 32 KB: one x row per thread
    __shared__ uint2 nodes[2][TB * M_NODES];       // 2 x 6048 B, double buffered
    __shared__ float leaves[2][TB * N_LEAF];       // 2 x 3072 B, double buffered
    // total 51 KB -> every LDS base is a compile-time offset < 64K, so the
    // ds_load 16-bit offset immediate absorbs all table bases.

    const int tid = threadIdx.x;
    const int blk = blockIdx.x;

    // ---- stage this block's 128 x-rows (async DMA, 16B per lane per op) ----
    {
        const float* gx = x + (size_t)blk * BLOCK * N_FEAT;
        unsigned xb = lds_off(xs);
        #pragma unroll
        for (int it = 0; it < N_FEAT / 4; ++it) {              // 16 iters
            int e = it * BLOCK + tid;                          // 16B chunk id
            async_ld_b128(xb + (unsigned)e * 16u, gx + e * 4);
        }
    }
    stage_trees(0, tid, feat, thresh, leaf, &nodes[0][0], &leaves[0][0]);
    wait_async0();
    __syncthreads();

    float acc0 = 0.f, acc1 = 0.f, acc2 = 0.f;
    // per-thread base of this sample's x row (byte address in LDS space)
    const char* xrowB = (const char*)xs + (unsigned)tid * (N_FEAT * 4);

    // Traversal state iB = nodeIdx*8 (byte offset into a tree's node table).
    // Node load: ds_load_b64 [iB + compile-time base]  -> zero addr VALU.
    // Update:    iB = 2*iB + (x > t ? 16 : 8)          -> cmp+cndmask+lshl_add.
    // Leaf:      ds_load_b32 [(iB>>1) + (base + t*256 - 252)].
    auto compute_batch = [&](const uint2* __restrict__ nd,
                             const float* __restrict__ lv) {
        const char* ndB = (const char*)nd;
        const char* lvB = (const char*)lv;
        #pragma unroll
        for (int j = 0; j < TB / N_CLASSES; ++j) {
            const unsigned o0 = (unsigned)(3 * j + 0) * (M_NODES * 8);
            const unsigned o1 = (unsigned)(3 * j + 1) * (M_NODES * 8);
            const unsigned o2 = (unsigned)(3 * j + 2) * (M_NODES * 8);
            unsigned iB0 = 0, iB1 = 0, iB2 = 0;
            #pragma unroll
            for (int d = 0; d < TREE_DEPTH; ++d) {
                uint2 n0 = *(const uint2*)(ndB + o0 + iB0);
                uint2 n1 = *(const uint2*)(ndB + o1 + iB1);
                uint2 n2 = *(const uint2*)(ndB + o2 + iB2);
                float xv0 = *(const float*)(xrowB + (n0.x << 2));
                float xv1 = *(const float*)(xrowB + (n1.x << 2));
                float xv2 = *(const float*)(xrowB + (n2.x << 2));
                // go_left (x <= t) -> child 2i+1 (+8B), else 2i+2 (+16B)
                iB0 = 2u * iB0 + ((xv0 > __uint_as_float(n0.y)) ? 16u : 8u);
                iB1 = 2u * iB1 + ((xv1 > __uint_as_float(n1.y)) ? 16u : 8u);
                iB2 = 2u * iB2 + ((xv2 > __uint_as_float(n2.y)) ? 16u : 8u);
            }
            // leaf byte = (idx-63)*4 = iB/2 - 252; the -252 and per-tree base
            // fold into the ds offset immediate.
            acc0 += *(const float*)(lvB + (unsigned)(3 * j + 0) * (N_LEAF * 4) - 252u + (iB0 >> 1));
            acc1 += *(const float*)(lvB + (unsigned)(3 * j + 1) * (N_LEAF * 4) - 252u + (iB1 >> 1));
            acc2 += *(const float*)(lvB + (unsigned)(3 * j + 2) * (N_LEAF * 4) - 252u + (iB2 >> 1));
        }
    };

    // 2x-unrolled batch loop: the ping/pong buffer is a compile-time constant,
    // so node/leaf table bases become ds_load offset immediates.
    for (int bb = 0; bb < NB; bb += 2) {
        // -------- even batch: compute buf0, prefetch (bb+1) -> buf1 --------
        stage_trees((bb + 1) * TB, tid, feat, thresh, leaf,
                    &nodes[1][0], &leaves[1][0]);               // bb+1 < NB always
        compute_batch(&nodes[0][0], &leaves[0][0]);
        wait_async0();
        __syncthreads();
        // -------- odd batch: compute buf1, prefetch (bb+2) -> buf0 --------
        if (bb + 2 < NB)
            stage_trees((bb + 2) * TB, tid, feat, thresh, leaf,
                        &nodes[0][0], &leaves[0][0]);
        compute_batch(&nodes[1][0], &leaves[1][0]);
        wait_async0();
        __syncthreads();
    }

    // ---- decision function + log_softmax over 3 classes ----
    float d0 = acc0 * LRATE, d1 = acc1 * LRATE, d2 = acc2 * LRATE;
    float m  = fmaxf(d0, fmaxf(d1, d2));
    float s  = expf(d0 - m) + expf(d1 - m) + expf(d2 - m);
    float l  = m + logf(s);
    size_t so = ((size_t)blk * BLOCK + tid) * 3;
    out[so + 0] = d0 - l;
    out[so + 1] = d1 - l;
    out[so + 2] = d2 - l;
}

extern "C" void kernel_launch(void* const* d_in, const int* in_sizes, int n_in,
                              void* d_out, int out_size, void* d_ws, size_t ws_size,
                              hipStream_t stream) {
    (void)in_sizes; (void)n_in; (void)out_size; (void)d_ws; (void)ws_size;
    const float* x      = (const float*)d_in[0];
    const int*   feat   = (const int*)  d_in[1];
    const float* thresh = (const float*)d_in[2];
    const float* leaf   = (const float*)d_in[3];
    float*       out    = (float*)d_out;

    dim3 grid(N_SAMPLES / BLOCK);   // 512 blocks
    dim3 block(BLOCK);              // 128 threads = 4 waves (wave32)
    gbt_eval_kernel<<<grid, block, 0, stream>>>(x, feat, thresh, leaf, out);
}